// QRN_23373212025361
// MI455X (gfx1250) — compile-verified
//
#include <hip/hip_runtime.h>
#include <cstdint>
#include <cstddef>

// ---------------------------------------------------------------------------
// QGRU for MI455X (gfx1250, wave32, WMMA).
//   I=H=64, T=2048, B=64, G=3H=192, 4I=4H=256, 4G=768.
//
// Phase 1 (build): materialize Hamilton-product weight matrices with the
//   gate permutation baked in:
//     Wih_cm, Whh_cm stored column-major [p][f], p = gate*256 + quat_out*64 + n
//   plus bias_perm[p] = b_ih + b_hh (permuted).
// Phase 2 (gi GEMM): gi[t,b,p] = x[t,b,:] @ Wih_cm[:,p] + bias_perm[p]
//   using v_wmma_f32_16x16x4_f32 (full fp32 precision; phase is HBM-bound).
// Phase 3 (scan): 4 persistent workgroups, W_hh gate-slices resident in LDS
//   (196KB) + h tile in LDS (65KB); cross-WG step sync via global semaphore.
// ---------------------------------------------------------------------------

typedef float v2f __attribute__((ext_vector_type(2)));
typedef float v8f __attribute__((ext_vector_type(8)));

#define T_LEN   2048
#define B_SZ    64
#define FOUR_H  256
#define FOUR_G  768
#define LDST    260            // LDS dword stride: %4==0 (b128 fill), K-pairs 8B aligned

// workspace layout (float offsets)
#define WS_WIH   0u            // 768*256 = 196608  (col-major [p][f])
#define WS_WHH   196608u       // 768*256
#define WS_BIAS  393216u       // 768
#define WS_HBUF  394240u       // 2 * 16384 (h double buffer)
#define WS_SEM   427008u       // semaphore (u32)
#define WS_GI    428032u       // 2048*64*768 = 100663296 floats
// total ≈ 101,091,328 floats ≈ 404 MB

__device__ __forceinline__ v8f wmma_f32(v2f a, v2f b, v8f c) {
    return __builtin_amdgcn_wmma_f32_16x16x4_f32(
        /*neg_a=*/false, a, /*neg_b=*/false, b,
        /*c_mod=*/(short)0, c, /*reuse_a=*/false, /*reuse_b=*/false);
}

// ---------------------------------------------------------------------------
// Phase 1: build permuted Hamilton matrices + fused bias.
// Hamilton block (row a = input quat, col b = output quat):
//   [[ r,  i,  j,  k], [-i, r, k, -j], [-j, -k, r, i], [-k, j, -i, r]]
// ---------------------------------------------------------------------------
__global__ void qgru_build_w(const float* __restrict__ wir, const float* __restrict__ wii,
                             const float* __restrict__ wij, const float* __restrict__ wik,
                             const float* __restrict__ whr, const float* __restrict__ whi,
                             const float* __restrict__ whj, const float* __restrict__ whk,
                             const float* __restrict__ b_ih, const float* __restrict__ b_hh,
                             float* __restrict__ Wih_cm, float* __restrict__ Whh_cm,
                             float* __restrict__ bias)
{
    int idx = blockIdx.x * 256 + threadIdx.x;          // 0 .. 2*196608-1
    if (idx >= 2 * 196608) return;
    const int which = idx / 196608;                    // 0 = ih, 1 = hh
    const int rem   = idx % 196608;
    const int p  = rem >> 8;                           // permuted column 0..767
    const int f  = rem & 255;                          // input feature 0..255
    const int g  = p >> 8;                             // gate 0..2
    const int qb = (p >> 6) & 3;                       // output quaternion
    const int n  = p & 63;
    const int qa = f >> 6;                             // input quaternion
    const int fi = f & 63;

    const int   comp[4][4] = {{0,1,2,3},{1,0,3,2},{2,3,0,1},{3,2,1,0}};
    const float sgn [4][4] = {{ 1.f, 1.f, 1.f, 1.f},
                              {-1.f, 1.f, 1.f,-1.f},
                              {-1.f,-1.f, 1.f, 1.f},
                              {-1.f, 1.f,-1.f, 1.f}};
    const int   cc = comp[qa][qb];
    const float s  = sgn [qa][qb];
    const int   wi = fi * 192 + g * 64 + n;            // (I,G) row-major

    if (which == 0) {
        const float* w = (cc == 0) ? wir : (cc == 1) ? wii : (cc == 2) ? wij : wik;
        Wih_cm[(size_t)p * 256 + f] = s * w[wi];
        if (f == 0) {
            const int ob = qb * 192 + g * 64 + n;      // original bias index
            bias[p] = b_ih[ob] + b_hh[ob];             // fold both biases into gi
        }
    } else {
        const float* w = (cc == 0) ? whr : (cc == 1) ? whi : (cc == 2) ? whj : whk;
        Whh_cm[(size_t)p * 256 + f] = s * w[wi];
    }
}

// ---------------------------------------------------------------------------
// Init: h0 <- hx, semaphore <- 0 (every launch; graph-replay safe)
// ---------------------------------------------------------------------------
__global__ void qgru_init(const float* __restrict__ hx,
                          float* __restrict__ hbuf,
                          unsigned int* __restrict__ sem)
{
    int i = blockIdx.x * 256 + threadIdx.x;
    if (i == 0) *sem = 0u;
    if (i < B_SZ * FOUR_H) hbuf[i] = hx[i];
}

// ---------------------------------------------------------------------------
// Phase 2: gi = x @ Wih + bias.  Block = 512 thr (16 waves).
// Block tile: 64 rows x 192 cols, K=256 staged once in LDS.
// Wave (mb = w&3, jb = w>>2) computes 16x16 tiles {jb, jb+4, jb+8}.
// ---------------------------------------------------------------------------
__global__ void __launch_bounds__(512)
qgru_gi_gemm(const float* __restrict__ x,
             const float* __restrict__ Wih_cm,
             const float* __restrict__ bias,
             float* __restrict__ gi)
{
    extern __shared__ float lds[];
    float* Alds = lds;                 // 64  x LDST  (x rows, K-major)
    float* Blds = lds + 64 * LDST;     // 192 x LDST  (W cols, K-major)

    const int tid = threadIdx.x;
    const unsigned lane = tid & 31u;
    const int wave = tid >> 5;
    const int mb = wave & 3;
    const int jb = wave >> 2;

    const int row0 = blockIdx.x * 64;      // of T*B = 131072
    const int col0 = blockIdx.y * 192;     // of 768

    for (int i = tid; i < 64 * 64; i += 512) {          // A: 64 rows x 64 float4
        const int r = i >> 6, q = i & 63;
        const float4 v = *(const float4*)(x + (size_t)(row0 + r) * 256 + 4 * q);
        *(float4*)(Alds + r * LDST + 4 * q) = v;
    }
    for (int i = tid; i < 192 * 64; i += 512) {         // B: 192 cols x 64 float4
        const int c = i >> 6, q = i & 63;
        const float4 v = *(const float4*)(Wih_cm + (size_t)(col0 + c) * 256 + 4 * q);
        *(float4*)(Blds + c * LDST + 4 * q) = v;
    }
    __syncthreads();

    v8f acc0 = {}, acc1 = {}, acc2 = {};
    const int arow = mb * 16 + (int)(lane & 15);
    const int khi  = (int)((lane >> 4) << 1);           // ISA A/B f32 frag striping
    const int cl0  = (jb + 0) * 16 + (int)(lane & 15);
    const int cl1  = (jb + 4) * 16 + (int)(lane & 15);
    const int cl2  = (jb + 8) * 16 + (int)(lane & 15);

#pragma unroll 8
    for (int k0 = 0; k0 < 256; k0 += 4) {
        const int k = k0 + khi;
        const v2f a  = *(const v2f*)(Alds + arow * LDST + k);
        const v2f b0 = *(const v2f*)(Blds + cl0  * LDST + k);
        const v2f b1 = *(const v2f*)(Blds + cl1  * LDST + k);
        const v2f b2 = *(const v2f*)(Blds + cl2  * LDST + k);
        acc0 = wmma_f32(a, b0, acc0);
        acc1 = wmma_f32(a, b1, acc1);
        acc2 = wmma_f32(a, b2, acc2);
    }

    const int hi8 = (int)((lane >> 4) << 3);
    const int n0 = col0 + (jb + 0) * 16 + (int)(lane & 15);
    const int n1 = col0 + (jb + 4) * 16 + (int)(lane & 15);
    const int n2 = col0 + (jb + 8) * 16 + (int)(lane & 15);
#pragma unroll
    for (int r = 0; r < 8; ++r) {
        const size_t row = (size_t)(row0 + mb * 16 + r + hi8);
        gi[row * FOUR_G + n0] = acc0[r] + bias[n0];
        gi[row * FOUR_G + n1] = acc1[r] + bias[n1];
        gi[row * FOUR_G + n2] = acc2[r] + bias[n2];
    }
}

// ---------------------------------------------------------------------------
// Phase 3: persistent scan. 4 WGs x 512 thr. WG c owns h columns [64c,64c+64).
// LDS-resident W_hh gate slices (3 x 256x64), h double-buffered in global ws.
// Wave (mb = w&3, jb = w>>2): one 16x16 tile per gate at (mb, jb);
// r/z/n land in the same lanes -> gate math fully in-register.
// ---------------------------------------------------------------------------
__global__ void __launch_bounds__(512)
qgru_scan(const float* __restrict__ gi,
          const float* __restrict__ Whh_cm,
          float* __restrict__ hbuf,          // 2 x (64*256)
          unsigned int* __restrict__ sem,
          float* __restrict__ out,           // (T,B,256)
          float* __restrict__ hlast)         // (B,256)
{
    extern __shared__ float lds[];
    float* Wlds = lds;                  // 192 cols x LDST (col-major, K inner)
    float* Hlds = lds + 192 * LDST;     // 64 rows  x LDST

    const int c   = blockIdx.x;         // 0..3: owned column slice
    const int tid = threadIdx.x;
    const unsigned lane = tid & 31u;
    const int wave = tid >> 5;
    const int mb = wave & 3;            // batch 16-block
    const int jb = wave >> 2;           // owned-col 16-block (0..3)

    // Resident W_hh slice: local col lc = gate*64 + j  <->  global p = gate*256 + c*64 + j
    for (int i = tid; i < 192 * 64; i += 512) {
        const int lc = i >> 6, q = i & 63;
        const int g = lc >> 6, j = lc & 63;
        const float4 v = *(const float4*)(Whh_cm + (size_t)(g * 256 + c * 64 + j) * 256 + 4 * q);
        *(float4*)(Wlds + lc * LDST + 4 * q) = v;
    }

    const int arow = mb * 16 + (int)(lane & 15);
    const int khi  = (int)((lane >> 4) << 1);
    const int jcol = jb * 16 + (int)(lane & 15);       // 0..63 within owned slice
    const int ng   = c * 64 + jcol;                    // global h column 0..255
    const int hi8  = (int)((lane >> 4) << 3);

    for (int t = 0; t < T_LEN; ++t) {
        if (tid == 0) {
            while (__hip_atomic_load(sem, __ATOMIC_ACQUIRE, __HIP_MEMORY_SCOPE_AGENT)
                   < 4u * (unsigned)t)
                __builtin_amdgcn_s_sleep(1);
        }
        __syncthreads();                                // also covers initial W fill

        const float* hsrc = hbuf + (size_t)(t & 1) * (B_SZ * FOUR_H);
        for (int i = tid; i < 64 * 64; i += 512) {      // full h -> LDS
            const int b = i >> 6, q = i & 63;
            const float4 v = *(const float4*)(hsrc + b * 256 + 4 * q);
            *(float4*)(Hlds + b * LDST + 4 * q) = v;
        }
        __syncthreads();

        v8f accR = {}, accZ = {}, accN = {};
#pragma unroll 8
        for (int k0 = 0; k0 < 256; k0 += 4) {
            const int k = k0 + khi;
            const v2f a  = *(const v2f*)(Hlds + arow * LDST + k);
            const v2f bR = *(const v2f*)(Wlds + (  0 + jcol) * LDST + k);
            const v2f bZ = *(const v2f*)(Wlds + ( 64 + jcol) * LDST + k);
            const v2f bN = *(const v2f*)(Wlds + (128 + jcol) * LDST + k);
            accR = wmma_f32(a, bR, accR);
            accZ = wmma_f32(a, bZ, accZ);
            accN = wmma_f32(a, bN, accN);
        }

        float* hdst = hbuf + (size_t)((t + 1) & 1) * (B_SZ * FOUR_H);
#pragma unroll
        for (int r = 0; r < 8; ++r) {
            const int b = mb * 16 + r + hi8;
            const size_t gbase = ((size_t)t * B_SZ + b) * FOUR_G;
            const float gr = gi[gbase + 0 * 256 + ng];
            const float gz = gi[gbase + 1 * 256 + ng];
            const float gn = gi[gbase + 2 * 256 + ng];
            const float hold = Hlds[b * LDST + ng];
            const float rg = 1.0f / (1.0f + __expf(-(gr + accR[r])));
            const float zg = 1.0f / (1.0f + __expf(-(gz + accZ[r])));
            const float nc = tanhf(gn + rg * accN[r]);
            const float hnew = nc + zg * (hold - nc);
            out[((size_t)t * B_SZ + b) * FOUR_H + ng] = hnew;
            hdst[b * FOUR_H + ng] = hnew;
            if (t == T_LEN - 1) hlast[b * FOUR_H + ng] = hnew;
        }
        __threadfence();
        __syncthreads();
        if (tid == 0)
            __hip_atomic_fetch_add(sem, 1u, __ATOMIC_RELEASE, __HIP_MEMORY_SCOPE_AGENT);
    }
}

// ---------------------------------------------------------------------------
extern "C" void kernel_launch(void* const* d_in, const int* in_sizes, int n_in,
                              void* d_out, int out_size, void* d_ws, size_t ws_size,
                              hipStream_t stream)
{
    (void)in_sizes; (void)n_in; (void)out_size; (void)ws_size;

    const float* x    = (const float*)d_in[0];
    const float* hx   = (const float*)d_in[1];
    const float* wir  = (const float*)d_in[2];
    const float* wii  = (const float*)d_in[3];
    const float* wij  = (const float*)d_in[4];
    const float* wik  = (const float*)d_in[5];
    const float* whr  = (const float*)d_in[6];
    const float* whi  = (const float*)d_in[7];
    const float* whj  = (const float*)d_in[8];
    const float* whk  = (const float*)d_in[9];
    const float* b_ih = (const float*)d_in[10];
    const float* b_hh = (const float*)d_in[11];

    float* ws   = (float*)d_ws;               // needs ~404 MB of workspace
    float* Wih  = ws + WS_WIH;
    float* Whh  = ws + WS_WHH;
    float* bias = ws + WS_BIAS;
    float* hbuf = ws + WS_HBUF;
    unsigned int* sem = (unsigned int*)(ws + WS_SEM);
    float* gi   = ws + WS_GI;

    float* out   = (float*)d_out;
    float* hlast = out + (size_t)T_LEN * B_SZ * FOUR_H;

    const size_t shmem = (size_t)(192 + 64) * LDST * sizeof(float);  // 266,240 B

    qgru_build_w<<<1536, 256, 0, stream>>>(wir, wii, wij, wik, whr, whi, whj, whk,
                                           b_ih, b_hh, Wih, Whh, bias);
    qgru_init<<<64, 256, 0, stream>>>(hx, hbuf, sem);

    dim3 ggrid((T_LEN * B_SZ) / 64, FOUR_G / 192);       // (2048, 4)
    qgru_gi_gemm<<<ggrid, 512, shmem, stream>>>(x, Wih, bias, gi);

    qgru_scan<<<4, 512, shmem, stream>>>(gi, Whh, hbuf, sem, out, hlast);
}